// DepthWiseConv2dImplicitGEMM_57251914055744
// MI455X (gfx1250) — compile-verified
//
#include <hip/hip_runtime.h>

typedef __attribute__((ext_vector_type(2))) float v2f;
typedef __attribute__((ext_vector_type(8))) float v8f;

#define DIM   384
#define HW    56
#define KS    31
#define PADR  15
#define FROWS 61          // padded filter rows: index (d - h_local + 30) in [0,60]
#define FCOLS 32          // 31 taps + 1 zero column pads K to 32
#define XW    96          // padded shared input row: w in [-15, 81)

// 8 K-chunks of 4 for one accumulator tile against one staged input row.
// A lane layout: (M = w_local, K = ko + v); B lane layout: (K = ko + v, N = h_local)
__device__ __forceinline__ void wmma_pass(v8f& acc, const float* __restrict__ fr,
                                          const float* __restrict__ xs, int ko) {
#pragma unroll
  for (int q = 0; q < 8; ++q) {
    const int ka = 4 * q + ko;
    v2f A = { xs[ka], xs[ka + 1] };
    v2f B = { fr[ka], fr[ka + 1] };
    acc = __builtin_amdgcn_wmma_f32_16x16x4_f32(
        /*neg_a=*/false, A, /*neg_b=*/false, B,
        /*c_mod=*/(short)0, acc, /*reuse_a=*/false, /*reuse_b=*/false);
  }
}

// Block = 4 waves. Wave i computes w-tile i (16 cols) x two stacked h-tiles
// (32 rows) of one (n, c) plane. Shared: zero-padded filter image + the
// double-buffered zero-padded input row.
// D[w,h] = sum_r sum_kw A_r[w,kw] * B_{r-ht}[kw,h],
//   A_r[w,kw] = X[r, w-15+kw],  B_d[kw,h] = F[d - h + 15, kw].
__global__ __launch_bounds__(128) void dwconv31_wmma_f32(
    const float* __restrict__ x, const float* __restrict__ wgt,
    const float* __restrict__ bias, float* __restrict__ out) {
  const int t    = threadIdx.x;
  const int lane = t & 31;
  const int wv   = t >> 5;     // wave id == w-tile index (0..3)
  const int lm   = lane & 15;  // N (= h_local) for B/C/D; M (= w_local) for A
  const int half = lane >> 4;  // K-pair select for A/B; M-block select for C/D

  const int hp = blockIdx.x;   // h pair: output rows hp*32 .. hp*32+31
  const int c  = blockIdx.y;
  const int n  = blockIdx.z;

  __shared__ float fpad[FROWS * FCOLS];   // zero-padded filter image (7.6 KB)
  __shared__ float xsh[2][XW];            // double-buffered padded input row
  __shared__ float otile[4][2][16][17];   // per-wave transpose buffers

  // ---- stage zero-padded filter once per block (shared by 8 tiles) ----
  for (int i = t; i < FROWS * FCOLS; i += 128) fpad[i] = 0.0f;
  __syncthreads();
  const float* wc = wgt + c * (KS * KS);
  for (int i = t; i < KS * KS; i += 128) {
    int kr = i / KS, kc = i - kr * KS;
    fpad[(kr + PADR) * FCOLS + kc] = wc[i];
  }
  __syncthreads();

  const int h0 = hp * 32;                       // upper tile rows h0..h0+15,
  const float* xc = x + (size_t)(n * DIM + c) * (HW * HW);  // lower: +16

  v8f acc0 = {}, acc1 = {};

  int rlo = h0 - PADR;  if (rlo < 0)      rlo = 0;
  int rhi = h0 + 46;    if (rhi > HW - 1) rhi = HW - 1;

  const int ko = 2 * half;
  const int xb = wv * 16 + lm;  // lane's A window base in the shared row

  for (int r = rlo; r <= rhi; ++r) {
    float* xr = xsh[r & 1];
    if (t < XW) {  // stage padded row: xr[i] = X[r, i-15]
      int wa = t - PADR;
      bool inb = (wa >= 0 && wa < HW);
      xr[t] = inb ? xc[r * HW + wa] : 0.0f;
      if (inb && r + 1 <= rhi)
        __builtin_prefetch(&xc[(r + 1) * HW + wa], 0, 3);  // global_prefetch_b8
    }
    __syncthreads();  // single barrier/row (double buffer covers WAR on xsh)

    const float* xs = &xr[xb];
    const int d0 = r - h0;             // upper-tile row offset, [-15, 46]
    if (r <= h0 + 30) {                // wave-uniform: EXEC stays full
      const float* fr0 = &fpad[(d0 - lm + 2 * PADR) * FCOLS];
      wmma_pass(acc0, fr0, xs, ko);
    }
    if (r >= h0 + 1) {                 // lower tile: d1 = d0 - 16 in [-15, 30]
      const float* fr1 = &fpad[(d0 - 16 - lm + 2 * PADR) * FCOLS];
      wmma_pass(acc1, fr1, xs, ko);
    }
  }

  // D lane layout: (M = v + 8*half, N = lm) == (w_local, h_local)
#pragma unroll
  for (int v = 0; v < 8; ++v) {
    otile[wv][0][v + 8 * half][lm] = acc0[v];
    otile[wv][1][v + 8 * half][lm] = acc1[v];
  }
  __syncthreads();

  const float bv = bias[c];
  float* oc = out + (size_t)(n * DIM + c) * (HW * HW);
  const int w0 = wv * 16;
#pragma unroll
  for (int j = 0; j < 8; ++j) {
    const int hh = 2 * j + half;  // lanes 0-15: row 2j, lanes 16-31: row 2j+1
    const int ww = w0 + lm;
    const int hA = h0 + hh;
    if (hA < HW && ww < HW) oc[hA * HW + ww] = otile[wv][0][lm][hh] + bv;
    const int hB = h0 + 16 + hh;
    if (hB < HW && ww < HW) oc[hB * HW + ww] = otile[wv][1][lm][hh] + bv;
  }
}

extern "C" void kernel_launch(void* const* d_in, const int* in_sizes, int n_in,
                              void* d_out, int out_size, void* d_ws, size_t ws_size,
                              hipStream_t stream) {
  const float* x    = (const float*)d_in[0];  // 32*384*56*56
  const float* wgt  = (const float*)d_in[1];  // 384*1*31*31
  const float* bias = (const float*)d_in[2];  // 384
  float* out = (float*)d_out;

  dim3 grid(2, DIM, 32);   // 2 h-pairs, 384 channels, 32 batch
  dim3 block(128);         // 4 waves: one 16-wide w-tile each
  dwconv31_wmma_f32<<<grid, block, 0, stream>>>(x, wgt, bias, out);
}